// GCN_22368189678004
// MI455X (gfx1250) — compile-verified
//
#include <hip/hip_runtime.h>
#include <hip/hip_bf16.h>

#define IN_DIM 128
#define HID    64

typedef float v2f __attribute__((ext_vector_type(2)));
typedef float v8f __attribute__((ext_vector_type(8)));

// ---------------------------------------------------------------------------
// Zero a float buffer
// ---------------------------------------------------------------------------
__global__ void k_zero(float* __restrict__ p, long long n) {
    long long i = (long long)blockIdx.x * blockDim.x + threadIdx.x;
    if (i < n) p[i] = 0.0f;
}

// ---------------------------------------------------------------------------
// deg[row[e]] += 1  (float atomics; counts <= 800000 are exact in f32)
// ---------------------------------------------------------------------------
__global__ void k_deg(const long long* __restrict__ ei, long long E,
                      float* __restrict__ deg) {
    long long e = (long long)blockIdx.x * blockDim.x + threadIdx.x;
    if (e < E) atomicAdd(&deg[ei[e]], 1.0f);
}

// ---------------------------------------------------------------------------
// norm[i] = rsqrt(1 + deg[i])  (in place)
// ---------------------------------------------------------------------------
__global__ void k_norm(float* __restrict__ norm, long long n) {
    long long i = (long long)blockIdx.x * blockDim.x + threadIdx.x;
    if (i < n) norm[i] = rsqrtf(1.0f + norm[i]);
}

// ---------------------------------------------------------------------------
// Dense GEMM h = x @ W + b via V_WMMA_F32_16X16X4_F32.
// One wave computes a 16(M) x 64(N) strip: 4 accumulators, K-loop of 32x K=4.
//
// Fragment layouts (wave32, per CDNA5 ISA 7.12.2):
//   A 16x4 f32 : lane M = lane&15, half = lane>>4; a[0]=A[M][k+2*half],
//                a[1]=A[M][k+2*half+1]
//   B 4x16 f32 : lane N = lane&15; b[0]=B[k+2*half][N], b[1]=B[k+2*half+1][N]
//   C 16x16 f32: c[v] = C[v + 8*half][lane&15]
// ---------------------------------------------------------------------------
__global__ void k_gemm_wmma(const float* __restrict__ x,
                            const float* __restrict__ W,
                            const float* __restrict__ bias,
                            float* __restrict__ h, int nrows16) {
    const int wave  = (int)((blockIdx.x * blockDim.x + threadIdx.x) >> 5);
    const int lane  = threadIdx.x & 31;
    const int mbase = wave * 16;
    if (mbase >= nrows16) return;           // whole-wave exit: EXEC stays full
    const int half = lane >> 4;
    const int l15  = lane & 15;

    v8f c[4];
#pragma unroll
    for (int t = 0; t < 4; ++t) {
        const float bb = bias[t * 16 + l15];
#pragma unroll
        for (int v = 0; v < 8; ++v) c[t][v] = bb;   // C init = bias broadcast
    }

    const float* xrow = x + (size_t)(mbase + l15) * IN_DIM;
#pragma unroll 4
    for (int k = 0; k < IN_DIM; k += 4) {
        v2f a;
        a.x = xrow[k + 2 * half + 0];
        a.y = xrow[k + 2 * half + 1];
#pragma unroll
        for (int t = 0; t < 4; ++t) {
            v2f b;
            b.x = W[(size_t)(k + 2 * half + 0) * HID + t * 16 + l15];
            b.y = W[(size_t)(k + 2 * half + 1) * HID + t * 16 + l15];
            c[t] = __builtin_amdgcn_wmma_f32_16x16x4_f32(
                false, a, false, b, (short)0, c[t], false, false);
        }
    }

#pragma unroll
    for (int t = 0; t < 4; ++t)
#pragma unroll
        for (int v = 0; v < 8; ++v)
            h[(size_t)(mbase + v + 8 * half) * HID + t * 16 + l15] = c[t][v];
}

// Scalar cleanup for rows not covered by full 16-row tiles (N=50000 has none,
// but keep it general). One thread per (row, col).
__global__ void k_gemm_tail(const float* __restrict__ x,
                            const float* __restrict__ W,
                            const float* __restrict__ bias,
                            float* __restrict__ h, int row0, int nrows) {
    int idx = blockIdx.x * blockDim.x + threadIdx.x;
    int row = row0 + (idx >> 6);
    int col = idx & 63;
    if (row >= nrows) return;
    float acc = bias[col];
    const float* xr = x + (size_t)row * IN_DIM;
    for (int k = 0; k < IN_DIM; ++k) acc += xr[k] * W[(size_t)k * HID + col];
    h[(size_t)row * HID + col] = acc;
}

// ---------------------------------------------------------------------------
// Sparse aggregation: agg[row[e]] += norm[col[e]] * h[col[e]]  (64 feats/edge)
// 16 lanes per edge, float4 chunk per lane -> lanes 0..15 read one edge's
// contiguous 256B (coalesced), then 4 f32 global atomics each.
// ---------------------------------------------------------------------------
__global__ void k_scatter(const long long* __restrict__ ei, long long E,
                          const float* __restrict__ h,
                          const float* __restrict__ norm,
                          float* __restrict__ agg) {
    long long gid   = (long long)blockIdx.x * blockDim.x + threadIdx.x;
    long long e     = gid >> 4;
    int       chunk = (int)(gid & 15);
    if (e >= E) return;
    long long r = ei[e];          // row  = edge_index[0][e]
    long long c = ei[E + e];      // col  = edge_index[1][e]
    float nc = norm[c];
    const float4 hv = *(const float4*)(h + c * HID + chunk * 4);
    float* dst = agg + r * HID + chunk * 4;
    atomicAdd(dst + 0, nc * hv.x);
    atomicAdd(dst + 1, nc * hv.y);
    atomicAdd(dst + 2, nc * hv.z);
    atomicAdd(dst + 3, nc * hv.w);
}

// ---------------------------------------------------------------------------
// h_next = norm*(agg + norm*h) = norm*agg + norm^2*h   (out may alias agg)
// ---------------------------------------------------------------------------
__global__ void k_update(float* __restrict__ out,
                         const float* __restrict__ agg,
                         const float* __restrict__ h,
                         const float* __restrict__ norm, long long total) {
    long long i = (long long)blockIdx.x * blockDim.x + threadIdx.x;
    if (i >= total) return;
    float nv = norm[i >> 6];
    out[i] = nv * agg[i] + nv * nv * h[i];
}

// ---------------------------------------------------------------------------
extern "C" void kernel_launch(void* const* d_in, const int* in_sizes, int n_in,
                              void* d_out, int out_size, void* d_ws, size_t ws_size,
                              hipStream_t stream) {
    const float*     x    = (const float*)d_in[0];
    const long long* ei   = (const long long*)d_in[1];   // int64 per reference
    const float*     W    = (const float*)d_in[2];
    const float*     bias = (const float*)d_in[3];
    float*           out  = (float*)d_out;

    const long long N = in_sizes[0] / IN_DIM;            // 50000
    const long long E = in_sizes[1] / 2;                 // 800000
    const long long NH = N * HID;

    // Workspace layout: norm[N] | bufA[N*HID] | bufB[N*HID]  (~25.8 MB)
    float* norm = (float*)d_ws;
    float* bufA = norm + N;
    float* bufB = bufA + NH;

    const int BS = 256;
    const int gN   = (int)((N + BS - 1) / BS);
    const int gE   = (int)((E + BS - 1) / BS);
    const int gNH  = (int)((NH + BS - 1) / BS);
    const int gSct = (int)((E * 16 + BS - 1) / BS);

    // degree -> norm
    k_zero<<<gN, BS, 0, stream>>>(norm, N);
    k_deg<<<gE, BS, 0, stream>>>(ei, E, norm);
    k_norm<<<gN, BS, 0, stream>>>(norm, N);

    // h = x @ W + b  (WMMA f32, 16-row tiles; 8 waves per 256-thread block)
    const int nrows16 = (int)((N / 16) * 16);
    const int waves   = nrows16 / 16;
    const int gGemm   = (waves + 7) / 8;
    k_gemm_wmma<<<gGemm, BS, 0, stream>>>(x, W, bias, bufA, nrows16);
    if (nrows16 < N) {
        int tail = (int)(N - nrows16) * HID;
        k_gemm_tail<<<(tail + BS - 1) / BS, BS, 0, stream>>>(x, W, bias, bufA,
                                                             nrows16, (int)N);
    }

    // Layer 0: h(bufA) -> bufB
    k_zero<<<gNH, BS, 0, stream>>>(bufB, NH);
    k_scatter<<<gSct, BS, 0, stream>>>(ei, E, bufA, norm, bufB);
    k_update<<<gNH, BS, 0, stream>>>(bufB, bufB, bufA, norm, NH);

    // Layer 1: h(bufB) -> bufA
    k_zero<<<gNH, BS, 0, stream>>>(bufA, NH);
    k_scatter<<<gSct, BS, 0, stream>>>(ei, E, bufB, norm, bufA);
    k_update<<<gNH, BS, 0, stream>>>(bufA, bufA, bufB, norm, NH);

    // Layer 2: h(bufA) -> d_out
    k_zero<<<gNH, BS, 0, stream>>>(bufB, NH);
    k_scatter<<<gSct, BS, 0, stream>>>(ei, E, bufA, norm, bufB);
    k_update<<<gNH, BS, 0, stream>>>(out, bufB, bufA, norm, NH);
}